// TensorTrainLayer_86818468921851
// MI455X (gfx1250) — compile-verified
//
#include <hip/hip_runtime.h>

// TT-layer forward on gfx1250 (MI455X), f32 end-to-end.
// Dominant contraction (step 1) runs on V_WMMA_F32_16X16X4_F32.
//
// Inputs (setup_inputs order): z (1024x4096 f32), core0 (1x16x16x32),
// core1 (32x16x16x32), core2 (32x16x16x1). Output: (1024, 4096) f32.
//
// d_ws usage: 1 MB for core1 rearranged into WMMA A-fragment order.

typedef float v2f __attribute__((ext_vector_type(2)));
typedef float v8f __attribute__((ext_vector_type(8)));

// ---------------------------------------------------------------------------
// Pre-pass: rearrange core1 (flat kk*512 + mm, kk=(r1,d), mm=(i,r2)) into
// per-lane WMMA A fragments:
//   rep[((Mt*128 + ks)*32 + L)] = { core1[(4ks+ko)*512 + Mt*16 + (L&15)],
//                                   core1[(4ks+ko+1)*512 + Mt*16 + (L&15)] }
// with ko = (L<16 ? 0 : 2)  -> matches 16x4 f32 A-matrix VGPR layout.
// ---------------------------------------------------------------------------
__global__ __launch_bounds__(256) void tt_prepass_core1(
    const float* __restrict__ core1, float* __restrict__ rep) {
  const int u  = blockIdx.x * 256 + threadIdx.x;  // 131072 total
  const int L  = u & 31;
  const int ks = (u >> 5) & 127;
  const int Mt = u >> 12;
  const int ko = 4 * ks + ((L >> 4) << 1);
  const int m  = Mt * 16 + (L & 15);
  v2f val;
  val.x = core1[ko * 512 + m];
  val.y = core1[(ko + 1) * 512 + m];
  ((v2f*)rep)[u] = val;
}

// ---------------------------------------------------------------------------
// Main kernel: one workgroup per (b, g). 256 threads = 8 waves (wave32).
// ---------------------------------------------------------------------------
__global__ __launch_bounds__(256) void tt_main(
    const float* __restrict__ z, const float* __restrict__ core0,
    const float* __restrict__ core2, const float* __restrict__ rep,
    float* __restrict__ out) {
  __shared__ __align__(16) float smem[16384];  // 64 KB, time-multiplexed
  float* zsh   = smem;         // [0,4096)   : z[b] as [c][d][e]
  float* w0sh  = smem + 4096;  // [4096,4608): core0[:,g,:] as [c][r1]
  float* fragB = smem + 4608;  // [4608,12800): uses 8192 floats: B fragments

  const int t    = threadIdx.x;
  const int lane = t & 31;
  const int wave = t >> 5;
  const int bid  = blockIdx.x;
  const int b    = bid >> 4;
  const int g    = bid & 15;

  // ---- stage z[b] (16 KB) and W0_g (2 KB) into LDS ----
  {
    const float4* zg = (const float4*)(z + b * 4096);
    float4* zs = (float4*)zsh;
#pragma unroll
    for (int i = 0; i < 4; ++i) zs[t + 256 * i] = zg[t + 256 * i];
#pragma unroll
    for (int i = 0; i < 2; ++i) {
      const int idx = t + 256 * i;  // idx = c*32 + r1
      w0sh[idx] = core0[(idx >> 5) * 512 + g * 32 + (idx & 31)];
    }
  }
  __syncthreads();

  // ---- phase A (VALU, 3% of FLOPs): A0_g[kk=(r1,d), e] -> B-fragment LDS ----
  {
    const int e     = t & 15;
    const int kbase = (t >> 4) * 32;
#pragma unroll 4
    for (int kk = kbase; kk < kbase + 32; ++kk) {
      const int r1 = kk >> 4, d = kk & 15;
      float acc = 0.f;
      const float* zp = zsh + d * 16 + e;
      const float* wp = w0sh + r1;
#pragma unroll
      for (int c = 0; c < 16; ++c) acc += zp[c * 256] * wp[c * 32];
      // scatter into B-fragment order (4x16 K-major tile per lane)
      const int ks = kk >> 2, r = kk & 3;
      const int Lw = ((r >> 1) << 4) + e;
      fragB[(ks * 32 + Lw) * 2 + (r & 1)] = acc;
    }
  }
  __syncthreads();

  // ---- phase B (94% of FLOPs): A1 = core1^T @ A0 via WMMA f32 16x16x4 ----
  // wave w owns M-tiles [4w, 4w+4); N = e (16); K = 512 in 128 steps of 4.
  const int mt0 = wave * 4;
  v8f acc[4];
#pragma unroll
  for (int j = 0; j < 4; ++j) acc[j] = (v8f){};

  const v2f* fb   = (const v2f*)fragB;
  const v2f* repv = (const v2f*)rep;
  const v2f* ap0  = repv + (size_t)mt0 * 4096 + lane;  // +j*4096 per M-tile

#pragma unroll 8
  for (int ks = 0; ks < 128; ++ks) {
    const v2f bf = fb[ks * 32 + lane];  // ds_load_b64, conflict-free
#pragma unroll
    for (int j = 0; j < 4; ++j) {
      const v2f af = ap0[j * 4096 + ks * 32];  // global_load_b64, coalesced (L2)
      acc[j] = __builtin_amdgcn_wmma_f32_16x16x4_f32(
          false, af, false, bf, (short)0, acc[j], false, false);
    }
  }
  __syncthreads();  // fragB dead; repurpose LDS

  float* A1s = smem;         // [0,8192)    : A1[(i,r2)=512][e=16]
  float* c2s = smem + 8192;  // [8192,16384): core2 flat (r2*16+e)*16 + k

  // ---- spill A1 tiles (C/D layout: VGPR v -> row v + 8*(lane>=16), col lane&15)
  {
    const int rowoff = (lane >> 4) * 8;
    const int col    = lane & 15;
#pragma unroll
    for (int j = 0; j < 4; ++j) {
      const int mt = mt0 + j;
#pragma unroll
      for (int v = 0; v < 8; ++v)
        A1s[(mt * 16 + rowoff + v) * 16 + col] = acc[j][v];
    }
  }
  // ---- stage core2 (32 KB) ----
  {
    const float4* cg = (const float4*)core2;
    float4* cs = (float4*)c2s;
#pragma unroll
    for (int i = 0; i < 8; ++i) cs[t + 256 * i] = cg[t + 256 * i];
  }
  __syncthreads();

  // ---- phase C (VALU, 3%): out[i,k] = sum_{r2,e} A1[(i,r2),e]*core2[r2,e,k]
  {
    const int i = t >> 4, k = t & 15;
    float o = 0.f;
    const float* ap = A1s + i * 32 * 16;
#pragma unroll 4
    for (int r2 = 0; r2 < 32; ++r2) {
      const float* a1r = ap + r2 * 16;
      const float* c2r = c2s + r2 * 256 + k;
#pragma unroll
      for (int e = 0; e < 16; ++e) o += a1r[e] * c2r[e * 16];
    }
    out[(size_t)b * 4096 + g * 256 + t] = o;  // coalesced
  }
}

// ---------------------------------------------------------------------------
extern "C" void kernel_launch(void* const* d_in, const int* in_sizes, int n_in,
                              void* d_out, int out_size, void* d_ws,
                              size_t ws_size, hipStream_t stream) {
  (void)in_sizes; (void)n_in; (void)out_size; (void)ws_size;
  const float* z     = (const float*)d_in[0];
  const float* core0 = (const float*)d_in[1];
  const float* core1 = (const float*)d_in[2];
  const float* core2 = (const float*)d_in[3];
  float* rep = (float*)d_ws;  // needs 1 MB scratch

  tt_prepass_core1<<<512, 256, 0, stream>>>(core1, rep);
  tt_main<<<16384, 256, 0, stream>>>(z, core0, core2, rep, (float*)d_out);
}